// GraphConvLayer_8048768713465
// MI455X (gfx1250) — compile-verified
//
#include <hip/hip_runtime.h>

typedef float v2f __attribute__((ext_vector_type(2)));
typedef float v8f __attribute__((ext_vector_type(8)));

#define O_NODES 50000
#define T_PREDS 200000
#define DIM     128
#define N_NODES (O_NODES + T_PREDS)

// Native fire-and-forget fp32 atomic add, device scope (coherent across WGPs).
// Avoids any chance of a CAS-loop fallback; tracked on STOREcnt, no return VGPR.
__device__ __forceinline__ void atomic_add_f32_dev(float* p, float v) {
    asm volatile("global_atomic_add_f32 %0, %1, off scope:SCOPE_DEV"
                 :: "v"(p), "v"(v)
                 : "memory");
}

// ---------------- degree / normalization ----------------

// deg[n] = 1 (self loop) + 1 if n < T (type-1 edge dst = t)
__global__ __launch_bounds__(256) void k_deg_init(float* __restrict__ deg) {
    int n = blockIdx.x * blockDim.x + threadIdx.x;
    if (n < N_NODES) deg[n] = (n < T_PREDS) ? 2.0f : 1.0f;
}

// type-2 edges: dst = edges[t,1]
__global__ __launch_bounds__(256) void k_deg_scatter(const int* __restrict__ edges,
                                                     float* __restrict__ deg) {
    int t = blockIdx.x * blockDim.x + threadIdx.x;
    if (t < T_PREDS) atomic_add_f32_dev(&deg[edges[2 * t + 1]], 1.0f);
}

__global__ __launch_bounds__(256) void k_dinv(const float* __restrict__ deg,
                                              float* __restrict__ dinv) {
    int n = blockIdx.x * blockDim.x + threadIdx.x;
    if (n < N_NODES) dinv[n] = rsqrtf(deg[n]);  // deg >= 1 always (self loops)
}

// ---------------- fp32 WMMA GEMM: xw = [obj;pred] @ W^T ----------------
// One wave computes a 16-row x 128-col stripe with 8 accumulators of
// v_wmma_f32_16x16x4_f32, K-loop over 128 in steps of 4.
//
// A (16x4, MxK) per-lane layout: m = lane&15, VGPR v holds K = (lane>>4)*2 + v
// B (4x16, KxN) per-lane layout: n = lane&15, VGPR v holds K = (lane>>4)*2 + v
// B[k][n] = W[n][k]  (since we need X @ W^T), so B loads are float2 from W rows.
// C/D (16x16 f32): lane half selects M block (0-7 / 8-15), n = lane&15.
__global__ __launch_bounds__(256) void k_gemm(const float* __restrict__ obj,
                                              const float* __restrict__ pred,
                                              const float* __restrict__ W,
                                              float* __restrict__ xw) {
    int gtid    = blockIdx.x * blockDim.x + threadIdx.x;
    int wave    = gtid >> 5;
    int lane    = threadIdx.x & 31;
    int rowBase = wave * 16;
    if (rowBase >= N_NODES) return;

    const float* X;
    int r0;
    if (rowBase < O_NODES) { X = obj;  r0 = rowBase; }            // O divisible by 16
    else                   { X = pred; r0 = rowBase - O_NODES; }

    const int m  = lane & 15;
    const int k0 = (lane >> 4) * 2;
    const float* arow = X + (size_t)(r0 + m) * DIM;

    v8f c0 = {0,0,0,0,0,0,0,0}, c1 = c0, c2 = c0, c3 = c0;
    v8f c4 = c0, c5 = c0, c6 = c0, c7 = c0;

    for (int kk = 0; kk < DIM; kk += 4) {
        v2f a = *(const v2f*)(arow + kk + k0);
        const float* wb = W + (size_t)kk + k0 + (size_t)m * DIM;
#define WMMA_STEP(CJ, J)                                                        \
        {                                                                       \
            v2f b = *(const v2f*)(wb + (size_t)(J) * 16 * DIM);                 \
            CJ = __builtin_amdgcn_wmma_f32_16x16x4_f32(false, a, false, b,      \
                                                       (short)0, CJ, false, false); \
        }
        WMMA_STEP(c0, 0) WMMA_STEP(c1, 1) WMMA_STEP(c2, 2) WMMA_STEP(c3, 3)
        WMMA_STEP(c4, 4) WMMA_STEP(c5, 5) WMMA_STEP(c6, 6) WMMA_STEP(c7, 7)
#undef WMMA_STEP
    }

    // store: lane half -> rows 0-7 or 8-15 of the 16-row block
    const int mrow = (lane >> 4) * 8;
    float* orow = xw + (size_t)rowBase * DIM + m;
#define STORE_TILE(CJ, J)                                                       \
    {                                                                           \
        _Pragma("unroll")                                                       \
        for (int v = 0; v < 8; ++v)                                             \
            orow[(size_t)(mrow + v) * DIM + (J) * 16] = CJ[v];                  \
    }
    STORE_TILE(c0, 0) STORE_TILE(c1, 1) STORE_TILE(c2, 2) STORE_TILE(c3, 3)
    STORE_TILE(c4, 4) STORE_TILE(c5, 5) STORE_TILE(c6, 6) STORE_TILE(c7, 7)
#undef STORE_TILE
}

// ---------------- combine: self loop + type-1 edge (atomic-free) ----------------
// out[n] = b + x[n]*dinv[n]^2 + (n < T ? x[edges[n,0]] * dinv[src]*dinv[n] : 0)
__global__ __launch_bounds__(256) void k_combine(const float* __restrict__ xw,
                                                 const float* __restrict__ dinv,
                                                 const int* __restrict__ edges,
                                                 const float* __restrict__ bias,
                                                 float* __restrict__ out) {
    int idx  = blockIdx.x * blockDim.x + threadIdx.x;
    int n    = idx >> 5;          // 32 lanes per row
    int lane = idx & 31;
    if (n >= N_NODES) return;
    int c = lane * 4;

    float4 xv = *(const float4*)(xw + (size_t)n * DIM + c);
    float4 bv = *(const float4*)(bias + c);
    float  dn = dinv[n];
    float  self = dn * dn;

    float4 acc;
    acc.x = bv.x + xv.x * self;
    acc.y = bv.y + xv.y * self;
    acc.z = bv.z + xv.z * self;
    acc.w = bv.w + xv.w * self;

    if (n < T_PREDS) {
        int   s  = edges[2 * n + 0];
        float w  = dinv[s] * dn;
        float4 sv = *(const float4*)(xw + (size_t)s * DIM + c);
        acc.x += sv.x * w;
        acc.y += sv.y * w;
        acc.z += sv.z * w;
        acc.w += sv.w * w;
    }
    *(float4*)(out + (size_t)n * DIM + c) = acc;
}

// ---------------- type-2 scatter: out[edges[t,1]] += x[t]*dinv[t]*dinv[dst] ----------------
__global__ __launch_bounds__(256) void k_scatter(const float* __restrict__ xw,
                                                 const float* __restrict__ dinv,
                                                 const int* __restrict__ edges,
                                                 float* __restrict__ out) {
    int idx  = blockIdx.x * blockDim.x + threadIdx.x;
    int t    = idx >> 5;
    int lane = idx & 31;
    if (t >= T_PREDS) return;
    int d = edges[2 * t + 1];
    float w = dinv[t] * dinv[d];
    int c = lane * 4;
    const float* src = xw + (size_t)t * DIM + c;
    float* dst = out + (size_t)d * DIM + c;
#pragma unroll
    for (int i = 0; i < 4; ++i) atomic_add_f32_dev(&dst[i], src[i] * w);
}

extern "C" void kernel_launch(void* const* d_in, const int* in_sizes, int n_in,
                              void* d_out, int out_size, void* d_ws, size_t ws_size,
                              hipStream_t stream) {
    const float* obj   = (const float*)d_in[0];
    const float* pred  = (const float*)d_in[1];
    const int*   edges = (const int*)d_in[2];
    const float* W     = (const float*)d_in[3];
    const float* bias  = (const float*)d_in[4];
    float* out = (float*)d_out;

    float* xw   = (float*)d_ws;                 // N*128 floats (128 MB)
    float* deg  = xw + (size_t)N_NODES * DIM;   // N floats
    float* dinv = deg + N_NODES;                // N floats

    k_deg_init<<<(N_NODES + 255) / 256, 256, 0, stream>>>(deg);
    k_deg_scatter<<<(T_PREDS + 255) / 256, 256, 0, stream>>>(edges, deg);
    k_dinv<<<(N_NODES + 255) / 256, 256, 0, stream>>>(deg, dinv);

    int nwaves = N_NODES / 16;                  // 15625 waves, 8 per block
    k_gemm<<<(nwaves * 32 + 255) / 256, 256, 0, stream>>>(obj, pred, W, xw);

    long long cthreads = (long long)N_NODES * 32;
    k_combine<<<(int)((cthreads + 255) / 256), 256, 0, stream>>>(xw, dinv, edges, bias, out);

    long long sthreads = (long long)T_PREDS * 32;
    k_scatter<<<(int)((sthreads + 255) / 256), 256, 0, stream>>>(xw, dinv, edges, out);
}